// FrequencyAwareBoundaryLoss_60988535603853
// MI455X (gfx1250) — compile-verified
//
#include <hip/hip_runtime.h>
#include <math.h>

// ---------------- problem constants ----------------
#define BB   64        // batch
#define HH   512       // height
#define WW   512       // width
#define RF   257       // rfft cols = W/2+1
#define WP   272       // RF padded to 17 tiles of 16
#define NB   16        // NUM_BINS
#define NIMG 128       // 2*BB images (pred-masked + gt-masked)
#define NTIL (32*17)   // 16x16 output tiles per image (544)

typedef __attribute__((ext_vector_type(2))) float v2f;
typedef __attribute__((ext_vector_type(8))) float v8f;

static __device__ __forceinline__ v8f wmma4(v2f a, v2f b, v8f c) {
  // D = A(16x4,f32) * B(4x16,f32) + C(16x16,f32), exact f32 on CDNA5 matrix pipe
  return __builtin_amdgcn_wmma_f32_16x16x4_f32(false, a, false, b, (short)0, c,
                                               false, false);
}

// ---------------- init kernels ----------------
__global__ void k_zero(float* counts) {
  if (threadIdx.x < NB) counts[threadIdx.x] = 0.0f;
}

// radial bin index table (matches jnp fftfreq/rfftfreq math in f32) + counts
__global__ void k_tables(int* binIdx, float* counts) {
  int i = blockIdx.x * blockDim.x + threadIdx.x;
  if (i >= HH * RF) return;
  int v = i / RF, u = i % RF;
  float fy = (float)(v < 256 ? v : v - 512) / 512.0f;
  float fx = (float)u / 512.0f;
  float r = sqrtf(fy * fy + fx * fx);
  float rmax = sqrtf(0.5f);            // max over grid, reached at (256,256)
  float rn = r / rmax;
  int b = (int)(rn * 15.0f);
  if (b > 15) b = 15;
  if (b < 0) b = 0;
  binIdx[i] = b;
  atomicAdd(&counts[b], 1.0f);         // integer-valued -> exact & deterministic
}

// DFT coefficient matrices. angle = 2*pi*((r*c) mod 512)/512 (exact reduction)
__global__ void k_dft(float* W1c, float* W1s, float* W2c, float* W2s) {
  int i = blockIdx.x * blockDim.x + threadIdx.x;
  if (i >= HH * HH) return;
  int r = i / HH, c = i % HH;
  int ph = (r * c) & 511;
  float ang = (float)ph * (3.14159265358979323846f / 256.0f);
  float s, co;
  sincosf(ang, &s, &co);
  W2c[i] = co;                         // column pass: cos
  W2s[i] = s;                          // column pass: sin
  if (c < WP) {
    float vc = 0.0f, vs = 0.0f;
    if (c < RF) { vc = co; vs = -s; }  // row pass: Re uses cos, Im uses -sin
    W1c[(size_t)r * WP + c] = vc;
    W1s[(size_t)r * WP + c] = vs;
  }
}

// ---------------- boundary + masked inputs ----------------
// separable 7-tap (k=7, pad 3, clipped SAME) row max/min pool of gt
__global__ void k_rowpool(const float* gt, float* rmx, float* rmn) {
  int i = blockIdx.x * blockDim.x + threadIdx.x;
  if (i >= BB * HH * WW) return;
  int w = i & 511;
  const float* row = gt + (i - w);
  int lo = w - 3 < 0 ? 0 : w - 3;
  int hi = w + 3 > 511 ? 511 : w + 3;
  float mx = -1e30f, mn = 1e30f;
  for (int x = lo; x <= hi; ++x) {
    float v = row[x];
    mx = fmaxf(mx, v);
    mn = fminf(mn, v);
  }
  rmx[i] = mx;
  rmn[i] = mn;
}

// column pass of the pools, boundary = clip(dil - ero, 0, 1),
// bmP = sigmoid(pred)*boundary, bmG = gt*boundary
__global__ void k_bm(const float* pred, const float* gt, const float* rmx,
                     const float* rmn, float* bmP, float* bmG) {
  int i = blockIdx.x * blockDim.x + threadIdx.x;
  if (i >= BB * HH * WW) return;
  int w = i & 511;
  int h = (i >> 9) & 511;
  int b = i >> 18;
  int base = (b << 18) + w;
  int lo = h - 3 < 0 ? 0 : h - 3;
  int hi = h + 3 > 511 ? 511 : h + 3;
  float mx = -1e30f, mn = 1e30f;
  for (int y = lo; y <= hi; ++y) {
    int j = base + (y << 9);
    mx = fmaxf(mx, rmx[j]);
    mn = fminf(mn, rmn[j]);
  }
  float bd = mx - mn;
  bd = bd < 0.0f ? 0.0f : (bd > 1.0f ? 1.0f : bd);
  float p = 1.0f / (1.0f + expf(-pred[i]));
  bmP[i] = p * bd;
  bmG[i] = gt[i] * bd;
}

// ---------------- stage 1: row DFT via WMMA ----------------
// s1r/s1i[img][h][u] = sum_w bm[img][h][w] * {cos,-sin}(2*pi*u*w/512)
__global__ void __launch_bounds__(32)
k_stage1(const float* bmP, const float* bmG, const float* W1c,
         const float* W1s, float* s1r, float* s1i) {
  int img = blockIdx.z;
  int row0 = blockIdx.y * 16;
  int col0 = blockIdx.x * 16;
  const float* X = (img < BB) ? (bmP + (size_t)img * HH * WW)
                              : (bmG + (size_t)(img - BB) * HH * WW);
  int lane = threadIdx.x;
  int m = lane & 15;
  int half2 = (lane >> 4) << 1;  // 0 or 2 (K offset of this half-wave)
  v8f aR = {0.f, 0.f, 0.f, 0.f, 0.f, 0.f, 0.f, 0.f};
  v8f aI = {0.f, 0.f, 0.f, 0.f, 0.f, 0.f, 0.f, 0.f};
  const float* arow = X + (size_t)(row0 + m) * WW;
  for (int k0 = 0; k0 < WW; k0 += 4) {
    int ka = k0 + half2;
    v2f a = *(const v2f*)(arow + ka);                    // A[m][ka], A[m][ka+1]
    const float* bp0 = W1c + (size_t)ka * WP + (col0 + m);
    const float* bp1 = W1s + (size_t)ka * WP + (col0 + m);
    v2f bc, bs;
    bc.x = bp0[0]; bc.y = bp0[WP];                       // B[ka][n], B[ka+1][n]
    bs.x = bp1[0]; bs.y = bp1[WP];
    aR = wmma4(a, bc, aR);
    aI = wmma4(a, bs, aI);
  }
  size_t obase = (size_t)img * HH * WP + (size_t)(col0 + m);
  int rbase = row0 + ((lane >> 4) << 3);
  for (int j = 0; j < 8; ++j) {
    size_t o = obase + (size_t)(rbase + j) * WP;
    s1r[o] = aR[j];
    s1i[o] = aI[j];
  }
}

// ---------------- stage 2: column DFT via WMMA + binned energy ----------------
// Re = C*Rr + S*Ri ; Im = C*Ri - S*Rr ; energy scaled by (1/(H*W))^2 (forward norm)
__global__ void __launch_bounds__(32)
k_stage2(const float* s1r, const float* s1i, const float* W2c,
         const float* W2s, const int* binIdx, float* partial) {
  int img = blockIdx.z;
  int v0 = blockIdx.y * 16;
  int u0 = blockIdx.x * 16;
  int lane = threadIdx.x;
  int m = lane & 15;
  int half2 = (lane >> 4) << 1;
  v8f re = {0.f, 0.f, 0.f, 0.f, 0.f, 0.f, 0.f, 0.f};
  v8f im = {0.f, 0.f, 0.f, 0.f, 0.f, 0.f, 0.f, 0.f};
  const float* crow = W2c + (size_t)(v0 + m) * HH;
  const float* srow = W2s + (size_t)(v0 + m) * HH;
  const float* br0 = s1r + (size_t)img * HH * WP + (u0 + m);
  const float* bi0 = s1i + (size_t)img * HH * WP + (u0 + m);
  for (int k0 = 0; k0 < HH; k0 += 4) {
    int ka = k0 + half2;
    v2f ac = *(const v2f*)(crow + ka);
    v2f as = *(const v2f*)(srow + ka);
    v2f br, bi;
    br.x = br0[(size_t)ka * WP]; br.y = br0[(size_t)(ka + 1) * WP];
    bi.x = bi0[(size_t)ka * WP]; bi.y = bi0[(size_t)(ka + 1) * WP];
    re = wmma4(ac, br, re);
    re = wmma4(as, bi, re);
    im = wmma4(ac, bi, im);
    v2f nbr; nbr.x = -br.x; nbr.y = -br.y;
    im = wmma4(as, nbr, im);
  }
  __shared__ float ener[256];
  __shared__ int   ebin[256];
  const float sc2 = (1.0f / (512.0f * 512.0f)) * (1.0f / (512.0f * 512.0f));
  int vbase = v0 + ((lane >> 4) << 3);
  int uu = u0 + m;
  for (int j = 0; j < 8; ++j) {
    float e = 0.0f;
    int b = 0;
    if (uu < RF) {
      float rr = re[j], ii = im[j];
      e = (rr * rr + ii * ii) * sc2;
      b = binIdx[(vbase + j) * RF + uu];
    }
    int slot = j * 32 + lane;
    ener[slot] = e;
    ebin[slot] = b;
  }
  __syncthreads();
  int tile = blockIdx.y * gridDim.x + blockIdx.x;  // 0..NTIL-1
  if (lane < NB) {                                  // fixed-order -> deterministic
    float acc = 0.0f;
    for (int t = 0; t < 256; ++t)
      if (ebin[t] == lane) acc += ener[t];
    partial[((size_t)img * NTIL + tile) * NB + lane] = acc;
  }
}

// deterministic per-(image,bin) reduction over tiles
__global__ void k_reduce(const float* partial, float* profile) {
  int i = blockIdx.x * blockDim.x + threadIdx.x;
  if (i >= NIMG * NB) return;
  int img = i / NB, b = i % NB;
  const float* p = partial + (size_t)img * NTIL * NB + b;
  float acc = 0.0f;
  for (int t = 0; t < NTIL; ++t) acc += p[(size_t)t * NB];
  profile[i] = acc;
}

// normalize profiles and compute weighted L1 mean
__global__ void __launch_bounds__(128)
k_final(const float* profile, const float* counts, const float* fw, float* out) {
  int tid = threadIdx.x;  // 0..127, one image each
  __shared__ float cnt[NB];
  __shared__ float fwv[NB];
  __shared__ float np[NIMG * NB];
  __shared__ float part[128];
  if (tid < NB) {
    cnt[tid] = fmaxf(counts[tid], 1.0f);
    fwv[tid] = fw[tid];
  }
  __syncthreads();
  float v[NB];
  float s = 0.0f;
  for (int b = 0; b < NB; ++b) {
    v[b] = profile[tid * NB + b] / cnt[b];
    s += v[b];
  }
  float inv = 1.0f / (s + 1e-6f);
  for (int b = 0; b < NB; ++b) np[tid * NB + b] = v[b] * inv;
  __syncthreads();
  float acc = 0.0f;
  if (tid < BB)
    for (int b = 0; b < NB; ++b)
      acc += fabsf(np[tid * NB + b] - np[(BB + tid) * NB + b]) * fwv[b];
  part[tid] = acc;
  __syncthreads();
  for (int st = 64; st > 0; st >>= 1) {
    if (tid < st) part[tid] += part[tid + st];
    __syncthreads();
  }
  if (tid == 0) out[0] = part[0] / (float)(BB * NB);
}

// ---------------- launcher ----------------
extern "C" void kernel_launch(void* const* d_in, const int* in_sizes, int n_in,
                              void* d_out, int out_size, void* d_ws,
                              size_t ws_size, hipStream_t stream) {
  const float* pred = (const float*)d_in[0];  // [64,1,512,512]
  const float* gt   = (const float*)d_in[1];  // [64,1,512,512]
  const float* fw   = (const float*)d_in[2];  // [16]
  float* out = (float*)d_out;

  // workspace layout (floats); row-pool buffers aliased under stage-1 spectra
  // (strictly ordered on one stream -> safe). total ~285 MB.
  float* ws = (float*)d_ws;
  const size_t HWt = (size_t)HH * WW;
  const size_t s1e = (size_t)NIMG * HH * WP;
  float* bmP  = ws;
  float* bmG  = bmP + (size_t)BB * HWt;
  float* regB = bmG + (size_t)BB * HWt;  // rowMax, later s1r
  float* regC = regB + s1e;              // rowMin, later s1i
  float* W1c  = regC + s1e;
  float* W1s  = W1c + (size_t)HH * WP;
  float* W2c  = W1s + (size_t)HH * WP;
  float* W2s  = W2c + (size_t)HH * HH;
  int*   binIdx  = (int*)(W2s + (size_t)HH * HH);
  float* counts  = (float*)(binIdx + (size_t)HH * RF);
  float* partial = counts + NB;
  float* profile = partial + (size_t)NIMG * NTIL * NB;

  k_zero<<<1, 32, 0, stream>>>(counts);
  {
    int n = HH * RF;
    k_tables<<<(n + 255) / 256, 256, 0, stream>>>(binIdx, counts);
  }
  {
    int n = HH * HH;
    k_dft<<<(n + 255) / 256, 256, 0, stream>>>(W1c, W1s, W2c, W2s);
  }
  {
    int n = BB * HH * WW;
    k_rowpool<<<(n + 255) / 256, 256, 0, stream>>>(gt, regB, regC);
    k_bm<<<(n + 255) / 256, 256, 0, stream>>>(pred, gt, regB, regC, bmP, bmG);
  }
  dim3 g1(WP / 16, HH / 16, NIMG);  // (17, 32, 128), one wave per tile
  k_stage1<<<g1, 32, 0, stream>>>(bmP, bmG, W1c, W1s, regB, regC);
  k_stage2<<<g1, 32, 0, stream>>>(regB, regC, W2c, W2s, binIdx, partial);
  {
    int n = NIMG * NB;
    k_reduce<<<(n + 255) / 256, 256, 0, stream>>>(partial, profile);
  }
  k_final<<<1, 128, 0, stream>>>(profile, counts, fw, out);
}